// WeightedGCN_46557445489259
// MI455X (gfx1250) — compile-verified
//
#include <hip/hip_runtime.h>
#include <hip/hip_bf16.h>

typedef __attribute__((ext_vector_type(2))) float v2f;
typedef __attribute__((ext_vector_type(8))) float v8f;

#define N_NODES 100000
#define N_EDGES 1600000
#define NUM_G   256
#define EMB     10
#define ZC      32
#define KC      64
#define KPAD    12      // EMB padded to multiple of 4 for 16x16x4 WMMA

// ---------------- fill: deg=2.0 (self-loop fill, improved=True), zero acc/sums/cnt
__global__ void gcn_fill(float* deg, float* acc, float* sums_cnt) {
    long i = (long)blockIdx.x * blockDim.x + threadIdx.x;
    if (i < N_NODES) deg[i] = 2.0f;
    if (i < (long)N_NODES * ZC) acc[i] = 0.0f;
    if (i < NUM_G * ZC + NUM_G) sums_cnt[i] = 0.0f;
}

// ---------------- degree: deg[col[e]] += w[e]
__global__ void gcn_degree(const int* __restrict__ col, const float* __restrict__ w,
                           float* deg) {
    int e = blockIdx.x * blockDim.x + threadIdx.x;
    if (e < N_EDGES) atomicAdd(&deg[col[e]], w[e]);
}

// ---------------- dinv = rsqrt(deg) (deg >= 2 always; keep reference guard)
__global__ void gcn_dinv(const float* __restrict__ deg, float* dinv) {
    int i = blockIdx.x * blockDim.x + threadIdx.x;
    if (i < N_NODES) {
        float d = deg[i];
        dinv[i] = d > 0.0f ? rsqrtf(d) : 0.0f;
    }
}

// ---------------- node transform: h[n,32] = embed_table[x[n]] @ W_conv via f32 WMMA
// one wave per 16-node tile; 100000/16 = 6250 tiles exactly; blockDim=64 (2 waves)
// -> 3125 blocks, every wave full, EXEC all-1s as WMMA requires.
// A: five unconditional b64 loads + branchless half-select (no exec divergence).
// B: W_conv staged in LDS, zero-padded 10x32 -> 12x32; unconditional ds reads.
__global__ void gcn_node_transform(const int* __restrict__ x,
                                   const float* __restrict__ emb,
                                   const float* __restrict__ Wc,
                                   float* __restrict__ h) {
    __shared__ float Wpad[KPAD * ZC];            // 1.5 KB of the 320 KB WGP LDS
    for (int i = threadIdx.x; i < KPAD * ZC; i += blockDim.x) {
        Wpad[i] = (i < EMB * ZC) ? Wc[i] : 0.0f; // row-major: i = k*32 + col
    }
    __syncthreads();

    int wave = (blockIdx.x * blockDim.x + threadIdx.x) >> 5;   // tile id 0..6249
    int lane = threadIdx.x & 31;
    int m    = lane & 15;           // node row within tile / column within N-tile
    int half = lane >> 4;           // 0: K pair {k,k+1}, 1: {k+2,k+3}
    int node = wave * 16 + m;
    int idx  = x[node];

    // Embedding row: 40-byte stride -> 8-byte aligned; 5 unconditional v2f loads.
    const v2f* erow2 = (const v2f*)(emb + (long)idx * EMB);
    v2f p0 = erow2[0], p1 = erow2[1], p2 = erow2[2], p3 = erow2[3], p4 = erow2[4];
    v2f zz = {0.0f, 0.0f};
    // A-operand K pairs per chunk, selected branchlessly by half.
    v2f avals[3];
    avals[0] = half ? p1 : p0;      // k = 0/2
    avals[1] = half ? p3 : p2;      // k = 4/6
    avals[2] = half ? zz : p4;      // k = 8/10 (10,11 are zero padding)

    v8f c0 = {}; v8f c1 = {};
    #pragma unroll
    for (int kc = 0; kc < 3; ++kc) {
        int k = kc * 4 + half * 2;  // < KPAD=12, always in-bounds in Wpad
        v2f b0, b1;
        b0.x = Wpad[k * ZC + m];
        b0.y = Wpad[(k + 1) * ZC + m];
        b1.x = Wpad[k * ZC + 16 + m];
        b1.y = Wpad[(k + 1) * ZC + 16 + m];
        c0 = __builtin_amdgcn_wmma_f32_16x16x4_f32(false, avals[kc], false, b0, (short)0, c0, false, false);
        c1 = __builtin_amdgcn_wmma_f32_16x16x4_f32(false, avals[kc], false, b1, (short)0, c1, false, false);
    }
    // D layout: VGPR v, lanes 0-15 -> M=v, lanes 16-31 -> M=v+8; N = m
    #pragma unroll
    for (int v = 0; v < 8; ++v) {
        int nrow = wave * 16 + v + half * 8;
        h[(long)nrow * ZC + m]      = c0[v];
        h[(long)nrow * ZC + 16 + m] = c1[v];
    }
}

// ---------------- edge scatter: one wave per edge, one lane per channel (Z=32=wave32)
__global__ void gcn_edge_scatter(const int* __restrict__ row, const int* __restrict__ col,
                                 const float* __restrict__ w, const float* __restrict__ dinv,
                                 const float* __restrict__ h, float* acc) {
    int e    = (blockIdx.x * blockDim.x + threadIdx.x) >> 5;
    int lane = threadIdx.x & 31;
    if (e >= N_EDGES) return;
    int r = row[e], c = col[e];
    float nrm = dinv[r] * w[e] * dinv[c];
    float v = nrm * h[(long)r * ZC + lane];           // coalesced 128B load
    atomicAdd(&acc[(long)c * ZC + lane], v);          // coalesced f32 atomics (L2-resident)
}

// ---------------- pool: fuse self-loop (norm = 2*dinv^2) + per-graph sums/counts
__global__ void gcn_pool(const int* __restrict__ batch, const float* __restrict__ acc,
                         const float* __restrict__ h, const float* __restrict__ dinv,
                         float* sums, float* cnt) {
    int node = (blockIdx.x * blockDim.x + threadIdx.x) >> 5;
    int lane = threadIdx.x & 31;
    if (node >= N_NODES) return;
    int g = batch[node];
    float di = dinv[node];
    float d2 = 2.0f * di * di;
    float v = acc[(long)node * ZC + lane] + d2 * h[(long)node * ZC + lane];
    atomicAdd(&sums[g * ZC + lane], v);
    if (lane == 0) atomicAdd(&cnt[g], 1.0f);
}

// ---------------- pooled[g,z] = (sums + cnt*b_conv) / max(cnt,1)
// (out = scatter + b_conv per node  =>  segsum(out) = segsum(scatter) + cnt*b_conv)
__global__ void gcn_finalize(const float* __restrict__ sums, const float* __restrict__ cnt,
                             const float* __restrict__ b_conv, float* pooled) {
    int i = blockIdx.x * blockDim.x + threadIdx.x;   // 256*32
    int g = i >> 5, z = i & 31;
    float c = cnt[g];
    pooled[i] = (sums[i] + c * b_conv[z]) / fmaxf(c, 1.0f);
}

// ---------------- logits = pooled[256,32] @ W_read[32,64] via f32 WMMA
// 16 waves (one 16-row tile each) in one 512-thread block -> EXEC all-1s.
__global__ void gcn_logits(const float* __restrict__ pooled, const float* __restrict__ Wr,
                           float* __restrict__ logits) {
    int wave = threadIdx.x >> 5;    // 0..15
    int lane = threadIdx.x & 31;
    int m    = lane & 15;
    int half = lane >> 4;
    int m0   = wave * 16;
    const float* arow = pooled + (m0 + m) * ZC;
    #pragma unroll
    for (int nt = 0; nt < 4; ++nt) {
        v8f c = {};
        #pragma unroll
        for (int kc = 0; kc < 8; ++kc) {        // K = 32
            int k = kc * 4 + half * 2;
            v2f a; a.x = arow[k]; a.y = arow[k + 1];
            v2f b;
            b.x = Wr[k * KC + nt * 16 + m];
            b.y = Wr[(k + 1) * KC + nt * 16 + m];
            c = __builtin_amdgcn_wmma_f32_16x16x4_f32(false, a, false, b, (short)0, c, false, false);
        }
        #pragma unroll
        for (int v = 0; v < 8; ++v)
            logits[(m0 + v + half * 8) * KC + nt * 16 + m] = c[v];
    }
}

extern "C" void kernel_launch(void* const* d_in, const int* in_sizes, int n_in,
                              void* d_out, int out_size, void* d_ws, size_t ws_size,
                              hipStream_t stream) {
    const int*   x      = (const int*)  d_in[0];                 // [N]
    const int*   eidx   = (const int*)  d_in[1];                 // [2,E]
    const float* ew     = (const float*)d_in[2];                 // [E]
    const int*   batch  = (const int*)  d_in[3];                 // [N]
    const float* emb    = (const float*)d_in[4];                 // [N,10]
    const float* Wc     = (const float*)d_in[5];                 // [10,32]
    const float* bc     = (const float*)d_in[6];                 // [32]
    const float* Wr     = (const float*)d_in[7];                 // [32,64]
    (void)in_sizes; (void)n_in; (void)out_size; (void)ws_size;

    const int* row = eidx;
    const int* col = eidx + N_EDGES;

    // workspace layout (floats): deg | dinv | h | acc | sums | cnt  (~26.5 MB)
    float* ws   = (float*)d_ws;
    float* deg  = ws;
    float* dinv = ws + (long)N_NODES;
    float* h    = ws + 2L  * N_NODES;
    float* acc  = ws + 34L * N_NODES;
    float* sums = ws + 66L * N_NODES;
    float* cnt  = sums + NUM_G * ZC;

    float* pooled_out = (float*)d_out;            // [256,32]
    float* logits_out = pooled_out + NUM_G * ZC;  // [256,64]

    // 1) init (covers acc region: 32*N threads)
    {
        long total = (long)N_NODES * ZC;
        int blocks = (int)((total + 255) / 256);
        gcn_fill<<<blocks, 256, 0, stream>>>(deg, acc, sums);
    }
    // 2) degree
    gcn_degree<<<(N_EDGES + 255) / 256, 256, 0, stream>>>(col, ew, deg);
    // 3) dinv
    gcn_dinv<<<(N_NODES + 255) / 256, 256, 0, stream>>>(deg, dinv);
    // 4) node transform: 6250 tiles, 2 waves/block of 64 -> 3125 blocks (exact)
    gcn_node_transform<<<3125, 64, 0, stream>>>(x, emb, Wc, h);
    // 5) edge scatter: 1.6M waves, 8/block -> 200000 blocks (exact)
    gcn_edge_scatter<<<N_EDGES / 8, 256, 0, stream>>>(row, col, ew, dinv, h, acc);
    // 6) pool: 100000 waves, 8/block -> 12500 blocks (exact)
    gcn_pool<<<N_NODES / 8, 256, 0, stream>>>(batch, acc, h, dinv, sums, cnt);
    // 7) finalize pooled: 8192 threads
    gcn_finalize<<<(NUM_G * ZC) / 256, 256, 0, stream>>>(sums, cnt, bc, pooled_out);
    // 8) logits: 16 waves in one 512-thread block
    gcn_logits<<<1, 512, 0, stream>>>(pooled_out, Wr, logits_out);
}